// GCN_42417097015629
// MI455X (gfx1250) — compile-verified
//
#include <hip/hip_runtime.h>
#include <math.h>

typedef float v2f __attribute__((ext_vector_type(2)));
typedef float v8f __attribute__((ext_vector_type(8)));

#define NSTRIDE 32   // row stride (floats) of all padded node-feature buffers

// ---------------------------------------------------------------------------
// Degree / normalization precompute
// ---------------------------------------------------------------------------
__global__ void k_fill(float* __restrict__ p, float val, int n) {
  int i = blockIdx.x * blockDim.x + threadIdx.x;
  if (i < n) p[i] = val;
}

__global__ void k_deg_edges(const int* __restrict__ dst, float* __restrict__ deg, int e) {
  int i = blockIdx.x * blockDim.x + threadIdx.x;
  if (i < e) unsafeAtomicAdd(&deg[dst[i]], 1.0f);
}

__global__ void k_dinv(float* __restrict__ deg, int n) {
  int i = blockIdx.x * blockDim.x + threadIdx.x;
  if (i < n) {
    float d = deg[i];
    deg[i] = (d > 0.0f) ? rsqrtf(d) : 0.0f;  // deg >= 1 always (self loop)
  }
}

__global__ void k_norm(const int* __restrict__ src, const int* __restrict__ dst,
                       const float* __restrict__ dinv, float* __restrict__ nrm, int e) {
  int i = blockIdx.x * blockDim.x + threadIdx.x;
  if (i < e) nrm[i] = dinv[src[i]] * dinv[dst[i]];
}

// ---------------------------------------------------------------------------
// Operand staging: zero-padded copies so the WMMA GEMM needs no masking.
// ---------------------------------------------------------------------------
__global__ void k_pad_x(const float* __restrict__ x, float* __restrict__ xp,
                        int n, int npad) {
  int i = blockIdx.x * blockDim.x + threadIdx.x;   // over npad*4
  if (i >= npad * 4) return;
  int node = i >> 2, c = i & 3;
  xp[i] = (node < n && c < 3) ? x[node * 3 + c] : 0.0f;
}

__global__ void k_pad_w(const float* __restrict__ w, float* __restrict__ wp,
                        int fin, int fout, int kp, int wstride) {
  int i = blockIdx.x * blockDim.x + threadIdx.x;   // over kp*wstride
  if (i >= kp * wstride) return;
  int r = i / wstride, c = i - r * wstride;
  wp[i] = (r < fin && c < fout) ? w[r * fout + c] : 0.0f;
}

__global__ void k_pad_vec(const float* __restrict__ v, float* __restrict__ vp,
                          int len, int padlen) {
  int i = blockIdx.x * blockDim.x + threadIdx.x;
  if (i < padlen) vp[i] = (i < len) ? v[i] : 0.0f;
}

// ---------------------------------------------------------------------------
// Dense node transform via V_WMMA_F32_16X16X4_F32.
// One wave per 16-row tile; operands pre-padded -> zero conditionals, EXEC
// all-ones throughout (WMMA requirement), fully unrolled K chain.
//   H:  padded input, row stride HSTRIDE, K cols [0,KP) valid (zeros beyond FIN)
//   Wp: padded weights, KP x (COLTILES*16), zeros outside [FIN,FOUT)
//   Out: padded stride-NSTRIDE buffer with npad rows
// ---------------------------------------------------------------------------
template <int KP, int HSTRIDE, int COLTILES, bool HASBIAS>
__global__ __launch_bounds__(256)
void k_gemm16(const float* __restrict__ H, const float* __restrict__ Wp,
              const float* __restrict__ biasp, float* __restrict__ Out,
              int nwaves) {
  constexpr int WSTRIDE = COLTILES * 16;
  int wv = blockIdx.x * 8 + (threadIdx.x >> 5);
  if (wv >= nwaves) return;              // wave-uniform exit
  int lane = threadIdx.x & 31;
  int hi   = lane >> 4;                  // 0: K pair {0,1}, 1: K pair {2,3}
  int m    = lane & 15;
  int row  = wv * 16 + m;                // A row (padded rows always valid)
  int col  = blockIdx.y * 16 + m;        // B/D column (padded cols always valid)

  v8f c;
  float bv = HASBIAS ? biasp[col] : 0.0f;
#pragma unroll
  for (int i = 0; i < 8; ++i) c[i] = bv;

  const float* hrow = H + row * HSTRIDE;
#pragma unroll
  for (int ks = 0; ks < KP / 4; ++ks) {
    int ka = ks * 4 + 2 * hi;            // even -> 8B-aligned float2 load
    v2f a = *reinterpret_cast<const v2f*>(hrow + ka);
    v2f b;
    b.x = Wp[ka * WSTRIDE + col];
    b.y = Wp[(ka + 1) * WSTRIDE + col];
    c = __builtin_amdgcn_wmma_f32_16x16x4_f32(false, a, false, b, (short)0, c,
                                              false, false);
  }

  float* obase = Out + (wv * 16 + 8 * hi) * NSTRIDE + col;
#pragma unroll
  for (int r = 0; r < 8; ++r) obase[r * NSTRIDE] = c[r];   // unconditional
}

template <int KP, int HSTRIDE, int COLTILES, bool HASBIAS>
static inline void gemm(const float* H, const float* Wp, const float* biasp,
                        float* out, int nwaves, hipStream_t stream) {
  dim3 grid((nwaves + 7) / 8, COLTILES);
  k_gemm16<KP, HSTRIDE, COLTILES, HASBIAS><<<grid, 256, 0, stream>>>(
      H, Wp, biasp, out, nwaves);
}

// ---------------------------------------------------------------------------
// Aggregation init: agg[n] = bias + dinv[n]^2 * hw[n] (self loop), and zero
// the next GEMM's padded K columns [F, WC).
// ---------------------------------------------------------------------------
template <int F, int WC>
__global__ void k_self_init(const float* __restrict__ hw, const float* __restrict__ dinv,
                            const float* __restrict__ bias, float* __restrict__ agg, int n) {
  int node = blockIdx.x * blockDim.x + threadIdx.x;
  if (node >= n) return;
  float w = dinv[node] * dinv[node];
  const float* hp = hw + node * NSTRIDE;
  float* ap = agg + node * NSTRIDE;
#pragma unroll
  for (int c = 0; c < F; ++c) ap[c] = bias[c] + w * hp[c];
#pragma unroll
  for (int c = F; c < WC; ++c) ap[c] = 0.0f;
}

template <int F>
__global__ void k_edge_agg(const int* __restrict__ src, const int* __restrict__ dst,
                           const float* __restrict__ nrm, const float* __restrict__ hw,
                           float* __restrict__ agg, int e) {
  int i = blockIdx.x * blockDim.x + threadIdx.x;
  if (i >= e) return;
  int s = src[i], d = dst[i];
  float w = nrm[i];
  const float* hp = hw + s * NSTRIDE;    // rows 128B aligned
  float vals[F];
#pragma unroll
  for (int c = 0; c + 4 <= F; c += 4) {
    float4 t = *reinterpret_cast<const float4*>(hp + c);
    vals[c] = t.x; vals[c + 1] = t.y; vals[c + 2] = t.z; vals[c + 3] = t.w;
  }
  if constexpr ((F & 3) == 2) {          // F = 6: trailing float2 (16B offset)
    float2 t = *reinterpret_cast<const float2*>(hp + (F & ~3));
    vals[F - 2] = t.x; vals[F - 1] = t.y;
  }
  float* ap = agg + d * NSTRIDE;
#pragma unroll
  for (int c = 0; c < F; ++c) unsafeAtomicAdd(&ap[c], w * vals[c]);
}

// ---------------------------------------------------------------------------
// Epilogue: MODE 0 = tanh, MODE 1 = l2norm then tanh, MODE 2 = l2norm only
// ---------------------------------------------------------------------------
template <int F, int MODE, int OSTRIDE>
__global__ void k_post(const float* __restrict__ in, float* __restrict__ out, int n) {
  int node = blockIdx.x * blockDim.x + threadIdx.x;
  if (node >= n) return;
  float v[F];
  const float* p = in + node * NSTRIDE;
  float ss = 0.0f;
#pragma unroll
  for (int c = 0; c < F; ++c) { v[c] = p[c]; ss += v[c] * v[c]; }
  if (MODE >= 1) {
    float inv = 1.0f / fmaxf(sqrtf(ss), 1e-12f);
#pragma unroll
    for (int c = 0; c < F; ++c) v[c] *= inv;
  }
  if (MODE <= 1) {
#pragma unroll
    for (int c = 0; c < F; ++c) v[c] = tanhf(v[c]);
  }
  float* q = out + node * OSTRIDE;
#pragma unroll
  for (int c = 0; c < F; ++c) q[c] = v[c];
}

// ---------------------------------------------------------------------------
extern "C" void kernel_launch(void* const* d_in, const int* in_sizes, int n_in,
                              void* d_out, int out_size, void* d_ws, size_t ws_size,
                              hipStream_t stream) {
  const float* x  = (const float*)d_in[0];
  const int*   ei = (const int*)d_in[1];
  const float* W1 = (const float*)d_in[2];
  const float* b1 = (const float*)d_in[3];
  const float* W2 = (const float*)d_in[4];
  const float* b2 = (const float*)d_in[5];
  const float* W3 = (const float*)d_in[6];
  const float* b3 = (const float*)d_in[7];
  const float* Wc = (const float*)d_in[8];
  const float* bc = (const float*)d_in[9];
  float* out = (float*)d_out;

  int n = in_sizes[0] / 3;
  int e = in_sizes[1] / 2;
  const int* src = ei;
  const int* dst = ei + e;
  int nwaves = (n + 15) / 16;
  int npad   = nwaves * 16;

  char* ws = (char*)d_ws;
  size_t off = 0;
  auto carve = [&](size_t bytes) -> float* {
    float* p = (float*)(ws + off);
    off += (bytes + 255) & ~(size_t)255;
    return p;
  };
  float* dinv = carve((size_t)n * 4);              // deg -> dinv (in place)
  float* nrm  = carve((size_t)e * 4);              // per-edge sym norm
  float* Xp   = carve((size_t)npad * 4 * 4);       // padded x [npad x 4]
  float* A    = carve((size_t)npad * NSTRIDE * 4); // hw buffer (gemm out)
  float* B    = carve((size_t)npad * NSTRIDE * 4); // agg / h buffer (gemm in)
  float* W1p  = carve(4 * 16 * 4);                 // padded weights
  float* W2p  = carve(8 * 16 * 4);
  float* W3p  = carve(12 * 32 * 4);
  float* Wcp  = carve(24 * 16 * 4);
  float* bcp  = carve(16 * 4);

  const int T = 256;
  int gn = (n + T - 1) / T;
  int ge = (e + T - 1) / T;

  // degrees + symmetric normalization (shared by all 3 conv layers)
  k_fill<<<gn, T, 0, stream>>>(dinv, 1.0f, n);     // self loop
  k_deg_edges<<<ge, T, 0, stream>>>(dst, dinv, e);
  k_dinv<<<gn, T, 0, stream>>>(dinv, n);
  k_norm<<<ge, T, 0, stream>>>(src, dst, dinv, nrm, e);

  // operand staging (all tiny except Xp)
  k_pad_x<<<(npad * 4 + T - 1) / T, T, 0, stream>>>(x, Xp, n, npad);
  k_pad_w<<<1, 256, 0, stream>>>(W1, W1p, 3, 6, 4, 16);
  k_pad_w<<<1, 256, 0, stream>>>(W2, W2p, 6, 12, 8, 16);
  k_pad_w<<<2, 256, 0, stream>>>(W3, W3p, 12, 24, 12, 32);
  k_pad_w<<<2, 256, 0, stream>>>(Wc, Wcp, 24, 13, 24, 16);
  k_pad_vec<<<1, 32, 0, stream>>>(bc, bcp, 13, 16);

  // ---- layer 1: 3 -> 6, tanh ----
  gemm<4, 4, 1, false>(Xp, W1p, nullptr, A, nwaves, stream);
  k_self_init<6, 8><<<gn, T, 0, stream>>>(A, dinv, b1, B, n);
  k_edge_agg<6><<<ge, T, 0, stream>>>(src, dst, nrm, A, B, e);
  k_post<6, 0, NSTRIDE><<<gn, T, 0, stream>>>(B, B, n);

  // ---- layer 2: 6 -> 12, tanh(l2norm(.)) ----
  gemm<8, NSTRIDE, 1, false>(B, W2p, nullptr, A, nwaves, stream);
  k_self_init<12, 12><<<gn, T, 0, stream>>>(A, dinv, b2, B, n);
  k_edge_agg<12><<<ge, T, 0, stream>>>(src, dst, nrm, A, B, e);
  k_post<12, 1, NSTRIDE><<<gn, T, 0, stream>>>(B, B, n);

  // ---- layer 3: 12 -> 24, l2norm ----
  gemm<12, NSTRIDE, 2, false>(B, W3p, nullptr, A, nwaves, stream);
  k_self_init<24, 24><<<gn, T, 0, stream>>>(A, dinv, b3, B, n);
  k_edge_agg<24><<<ge, T, 0, stream>>>(src, dst, nrm, A, B, e);
  k_post<24, 2, NSTRIDE><<<gn, T, 0, stream>>>(B, B, n);

  // ---- classifier: 24 -> 13 (+bias), l2norm -> d_out [N x 13] ----
  gemm<24, NSTRIDE, 1, true>(B, Wcp, bcp, A, nwaves, stream);
  k_post<13, 2, 13><<<gn, T, 0, stream>>>(A, out, n);
}